// LoRALayer_27255862460738
// MI455X (gfx1250) — compile-verified
//
#include <hip/hip_runtime.h>
#include <hip/hip_bf16.h>

// LoRA forward for MI455X (gfx1250, wave32, WMMA).
// out[M,N] = orig[M,N] + 2.0 * (x[M,K] @ A[K,R]) @ B[R,N],  M=16384 K=4096 R=8 N=4096, fp32.
// Memory-bound (~770 MB @ 23.3 TB/s ~ 33us floor): fp32 end-to-end via
// V_WMMA_F32_16X16X4_F32 (fp32 matrix pipe), non-temporal hints on the 256MB
// streaming tensors (x / orig / out) so the 192MB L2 keeps lora_A/lora_B/low hot.

typedef __attribute__((ext_vector_type(2))) float v2f;
typedef __attribute__((ext_vector_type(8))) float v8f;

constexpr int M_ROWS = 4 * 4096;   // B*S = 16384
constexpr int K_IN   = 4096;       // D_IN
constexpr int N_OUT  = 4096;       // D_OUT
constexpr int RANK   = 8;
constexpr float SCALING = 2.0f;

// ---------------------------------------------------------------------------
// Kernel 1: low[M, RANK] = x[M, K] @ A[K, RANK]
// One wave per 16-row tile of x. 16x16x4 f32 WMMA, K looped in steps of 4.
// A-operand (x, 16x4):  lane (half,sub) -> row m0+sub; v0 = K=k0+2*half, v1 = +1.
// B-operand (A, 4x16):  lane column = sub; v0 = row K=k0+2*half, v1 = +1.
//   Rank is 8 < 16: lanes sub>=8 clamp their column index to 0 (branch-free,
//   in-bounds). Their accumulator columns compute duplicate garbage that is
//   simply never stored -- GEMM columns are independent, so this is exact for
//   the 8 real columns and keeps EXEC untouched in the hot loop (WMMA needs
//   EXEC all-ones, and the EXEC save/restore per iteration was pure overhead).
// C/D 16x16 f32: VGPR g -> row (g + 8*half), col = sub.
// ---------------------------------------------------------------------------
__global__ void __launch_bounds__(256) lora_down_wmma(
    const float* __restrict__ x,
    const float* __restrict__ Amat,
    float* __restrict__ low)
{
    const unsigned wave = threadIdx.x >> 5;
    const unsigned lane = threadIdx.x & 31u;
    const unsigned half = lane >> 4;     // 0: lanes 0-15, 1: lanes 16-31
    const unsigned sub  = lane & 15u;

    const unsigned tile = blockIdx.x * (blockDim.x >> 5) + wave; // 16-row tile
    const unsigned m0   = tile * 16u;

    const float* xrow = x + (size_t)(m0 + sub) * K_IN;
    const unsigned bcol = (sub < (unsigned)RANK) ? sub : 0u;  // clamp, stay in-bounds

    v8f acc = {};
    for (unsigned k0 = 0; k0 < (unsigned)K_IN; k0 += 4) {
        const unsigned ka = k0 + 2u * half;        // even -> 8B aligned
        v2f a = __builtin_nontemporal_load((const v2f*)(xrow + ka));

        v2f b;
        b.x = Amat[(size_t)ka * RANK + bcol];
        b.y = Amat[(size_t)(ka + 1) * RANK + bcol];

        acc = __builtin_amdgcn_wmma_f32_16x16x4_f32(
            /*neg_a=*/false, a, /*neg_b=*/false, b,
            /*c_mod=*/(short)0, acc, /*reuse_a=*/false, /*reuse_b=*/false);
    }

    if (sub < (unsigned)RANK) {
        #pragma unroll
        for (int g = 0; g < 8; ++g) {
            const unsigned row = m0 + (unsigned)g + 8u * half;
            low[(size_t)row * RANK + sub] = acc[g];
        }
    }
}

// ---------------------------------------------------------------------------
// Kernel 2: out[M,N] = orig[M,N] + SCALING * (low[M,RANK] @ Bmat[RANK,N])
// One wave per 16x16 output tile; K=RANK=8 -> two 16x16x4 f32 WMMAs, then the
// fused bandwidth-bound epilogue with NT load/store (no reuse on orig/out).
// ---------------------------------------------------------------------------
__global__ void __launch_bounds__(256) lora_up_add_wmma(
    const float* __restrict__ low,
    const float* __restrict__ Bmat,
    const float* __restrict__ orig,
    float* __restrict__ out)
{
    const unsigned wave = threadIdx.x >> 5;
    const unsigned lane = threadIdx.x & 31u;
    const unsigned half = lane >> 4;
    const unsigned sub  = lane & 15u;

    constexpr unsigned TILES_N = N_OUT / 16;   // 256
    const unsigned tile = blockIdx.x * (blockDim.x >> 5) + wave;
    const unsigned tm = tile / TILES_N;
    const unsigned tn = tile % TILES_N;
    const unsigned m0 = tm * 16u;
    const unsigned n0 = tn * 16u;

    const float* lrow = low + (size_t)(m0 + sub) * RANK;

    v8f acc = {};
    #pragma unroll
    for (unsigned s = 0; s < 2; ++s) {
        const unsigned ka = s * 4u + 2u * half;    // 0..7, even -> aligned
        v2f a = *(const v2f*)(lrow + ka);

        v2f b;
        b.x = Bmat[(size_t)ka * N_OUT + n0 + sub];
        b.y = Bmat[(size_t)(ka + 1) * N_OUT + n0 + sub];

        acc = __builtin_amdgcn_wmma_f32_16x16x4_f32(
            false, a, false, b, (short)0, acc, false, false);
    }

    #pragma unroll
    for (int g = 0; g < 8; ++g) {
        const size_t idx = (size_t)(m0 + (unsigned)g + 8u * half) * N_OUT + n0 + sub;
        const float o = __builtin_nontemporal_load(orig + idx);
        __builtin_nontemporal_store(o + SCALING * acc[g], out + idx);
    }
}

// ---------------------------------------------------------------------------
extern "C" void kernel_launch(void* const* d_in, const int* in_sizes, int n_in,
                              void* d_out, int out_size, void* d_ws, size_t ws_size,
                              hipStream_t stream)
{
    const float* x    = (const float*)d_in[0];  // [4,4096,4096]
    const float* orig = (const float*)d_in[1];  // [4,4096,4096]
    const float* Amat = (const float*)d_in[2];  // [4096,8]
    const float* Bmat = (const float*)d_in[3];  // [8,4096]
    float* out = (float*)d_out;
    float* low = (float*)d_ws;                  // [16384,8] f32 = 512 KB scratch

    const dim3 blk(256);                        // 8 waves/block (wave32)

    // Kernel 1: 16384 rows / (16 rows/wave * 8 waves/block) = 128 blocks
    lora_down_wmma<<<dim3(M_ROWS / (16 * 8)), blk, 0, stream>>>(x, Amat, low);

    // Kernel 2: (16384/16)*(4096/16) = 262144 tiles / 8 waves = 32768 blocks
    lora_up_add_wmma<<<dim3((M_ROWS / 16) * (N_OUT / 16) / 8), blk, 0, stream>>>(
        low, Bmat, orig, out);
}